// RNN_41085657153750
// MI455X (gfx1250) — compile-verified
//
#include <hip/hip_runtime.h>
#include <hip/hip_bf16.h>
#include <math.h>

// Problem sizes (match the reference).
#define SEQ   8192
#define ISZ   512
#define HSZ   2048
#define OSZ   512

// Recurrent phase: persistent workgroups, W rows cached in LDS.
#define NWG_B   128
#define ROWS_B  (HSZ / NWG_B)     // 16 rows of W per workgroup
#define WSTR    2064              // padded LDS row stride (floats): 2064%64==16 -> conflict-free

typedef __attribute__((ext_vector_type(2))) float v2f;
typedef __attribute__((ext_vector_type(8))) float v8f;

// gfx1250 async global->LDS builtin operand types:
// param0 = global (AS1 / __device__) pointer to int __vector(4)
// param1 = LDS (AS3) pointer to int __vector(4)
typedef int v4i __attribute__((vector_size(16)));
#define GPTR4(p) ((__attribute__((address_space(1))) v4i*)(p))
#define LPTR4(p) ((__attribute__((address_space(3))) v4i*)(p))

// ---------------------------------------------------------------------------
// Phase A: Xu = X @ U^T + U_b   ([8192,512] x [2048,512]^T -> [8192,2048])
// f32 WMMA 16x16x4. Block = 256 thr = 8 waves; each wave owns one 16x16 C tile.
// Grid: (SEQ/16, HSZ/128). A-tile layout: lane<16 -> K=0,1 ; lane>=16 -> K=2,3.
// ---------------------------------------------------------------------------
__global__ void __launch_bounds__(256)
xu_gemm(const float* __restrict__ x, const float* __restrict__ Uw,
        const float* __restrict__ Ub, float* __restrict__ xu)
{
    const int wave = threadIdx.x >> 5;
    const int lane = threadIdx.x & 31;
    const int r0 = blockIdx.x * 16;                 // sequence rows
    const int c0 = blockIdx.y * 128 + wave * 16;    // hidden cols for this wave
    const int m  = lane & 15;                       // A row index / B col index
    const int ks = (lane >> 4) << 1;                // K sub-offset: 0 or 2

    const float* arow = x  + (size_t)(r0 + m) * ISZ + ks;   // A[m][ks..ks+1]
    const float* brow = Uw + (size_t)(c0 + m) * ISZ + ks;   // B[ks..][n], n==m

    v8f acc = {0.f, 0.f, 0.f, 0.f, 0.f, 0.f, 0.f, 0.f};
    #pragma unroll 8
    for (int k0 = 0; k0 < ISZ; k0 += 4) {
        v2f a = *(const v2f*)(arow + k0);
        v2f b = *(const v2f*)(brow + k0);
        acc = __builtin_amdgcn_wmma_f32_16x16x4_f32(
                  false, a, false, b, (short)0, acc, false, false);
    }

    // C/D layout: VGPR i -> row = i + (lane>=16 ? 8 : 0), col = lane&15
    const int rbase = r0 + ((lane >> 4) << 3);
    const int c     = c0 + m;
    const float bias = Ub[c];
    #pragma unroll
    for (int i = 0; i < 8; ++i)
        xu[(size_t)(rbase + i) * HSZ + c] = acc[i] + bias;
}

// ---------------------------------------------------------------------------
// Phase B: sequential recurrence. 128 persistent WGs; WG b owns W rows
// [16b, 16b+16) resident in LDS for all 8192 steps (zero W re-reads from the
// memory hierarchy on the critical path). h is double-buffered in global
// scratch; per-step device barrier = atomic arrive counter + one broadcast
// flag written by the last arriver. Async global->LDS copies stage both the
// W tile (once) and h_{t-1} (each step) without a VGPR round-trip.
// ---------------------------------------------------------------------------
__global__ void __launch_bounds__(256)
rnn_recur(const float* __restrict__ Wmat, const float* __restrict__ Wb,
          float* __restrict__ xuH, float* __restrict__ hb,
          unsigned* __restrict__ counter, unsigned* __restrict__ flag)
{
    __shared__ __align__(16) float Wl[ROWS_B * WSTR];   // 129 KB: 16 W rows, padded
    __shared__ __align__(16) float hl[HSZ];             // 8 KB: staged h_{t-1}

    const int r0 = blockIdx.x * ROWS_B;

    // One-time async preload of this WG's 16 rows of W into LDS (b128 chunks).
    {
        const int f0 = threadIdx.x * 8;                 // 8 floats = 32 B per thread/row
        #pragma unroll 4
        for (int j = 0; j < ROWS_B; ++j) {
            const float* src = Wmat + (size_t)(r0 + j) * HSZ + f0;
            float*       dst = Wl + j * WSTR + f0;
            __builtin_amdgcn_global_load_async_to_lds_b128(GPTR4(src),     LPTR4(dst),     0, 0);
            __builtin_amdgcn_global_load_async_to_lds_b128(GPTR4(src + 4), LPTR4(dst + 4), 0, 0);
        }
    }

    const int j = threadIdx.x >> 4;   // row 0..15 (wave w covers rows 2w, 2w+1)
    const int p = threadIdx.x & 15;   // 16 lanes cooperate per row
    const float wb = Wb[r0 + j];

    __builtin_amdgcn_s_wait_asynccnt(0);
    __syncthreads();

    for (int t = 0; t < SEQ; ++t) {
        // Async-stage h_{t-1} into LDS: 256 threads x 2 x b128 = 8 KB.
        {
            const float* hprev = hb + (t & 1) * HSZ;
            const float* src = hprev + threadIdx.x * 4;
            float*       dst = hl + threadIdx.x * 4;
            __builtin_amdgcn_global_load_async_to_lds_b128(GPTR4(src),        LPTR4(dst),        0, 0);
            __builtin_amdgcn_global_load_async_to_lds_b128(GPTR4(src + 1024), LPTR4(dst + 1024), 0, 0);
        }

        // Hoisted: overlap the Xu[t] load with the dot product below.
        const size_t o = (size_t)t * HSZ + r0 + j;
        float xuv = 0.f;
        if (p == 0) xuv = xuH[o];

        __builtin_amdgcn_s_wait_asynccnt(0);
        __syncthreads();

        // 16-lane strided dot product; stride 16 + WSTR padding => no bank conflicts.
        float sum = 0.f;
        const float* wr = Wl + j * WSTR;
        #pragma unroll 8
        for (int i = 0; i < HSZ / 16; ++i) {
            const int k = p + (i << 4);
            sum = fmaf(wr[k], hl[k], sum);
        }
        #pragma unroll
        for (int off = 8; off; off >>= 1) sum += __shfl_xor(sum, off, 16);

        if (p == 0) {
            const float v = tanhf(xuv + wb + sum);
            xuH[o] = v;                              // H[t] replaces Xu[t]
            hb[((t + 1) & 1) * HSZ + r0 + j] = v;    // next h buffer
        }

        if (t == SEQ - 1) break;                     // no consumer of the last barrier

        // Device-wide barrier: arrive on counter, last arriver broadcasts t+1.
        __threadfence();
        __syncthreads();
        if (threadIdx.x == 0) {
            const unsigned tgt = (unsigned)NWG_B * (unsigned)(t + 1);
            const unsigned old = __hip_atomic_fetch_add(counter, 1u, __ATOMIC_ACQ_REL,
                                                        __HIP_MEMORY_SCOPE_AGENT);
            if (old + 1u == tgt) {
                __hip_atomic_store(flag, (unsigned)(t + 1), __ATOMIC_RELEASE,
                                   __HIP_MEMORY_SCOPE_AGENT);
            } else {
                while (__hip_atomic_load(flag, __ATOMIC_ACQUIRE,
                                         __HIP_MEMORY_SCOPE_AGENT) < (unsigned)(t + 1))
                    __builtin_amdgcn_s_sleep(2);
            }
        }
        __syncthreads();
        __threadfence();
    }
}

// ---------------------------------------------------------------------------
// Phase C: logits = H @ V^T + V_b, then fused row-wise log_softmax.
// Block handles a 16x512 output tile (one 16-row time block). 8 waves x 4
// WMMA col-tiles each = 512 cols. Epilogue: tile -> LDS, logsumexp, write.
// ---------------------------------------------------------------------------
__global__ void __launch_bounds__(256)
out_gemm_lsm(const float* __restrict__ Hm, const float* __restrict__ Vw,
             const float* __restrict__ Vb, float* __restrict__ out)
{
    __shared__ float lg[16 * (OSZ + 4)];    // 16 x 516 (padded) logits tile

    const int wave = threadIdx.x >> 5;
    const int lane = threadIdx.x & 31;
    const int t0 = blockIdx.x * 16;
    const int m  = lane & 15;
    const int ks = (lane >> 4) << 1;

    const float* arow = Hm + (size_t)(t0 + m) * HSZ + ks;
    const float* brow[4];
    #pragma unroll
    for (int w = 0; w < 4; ++w)
        brow[w] = Vw + (size_t)(wave * 64 + w * 16 + m) * HSZ + ks;

    v8f acc[4];
    #pragma unroll
    for (int w = 0; w < 4; ++w) acc[w] = (v8f){0.f,0.f,0.f,0.f,0.f,0.f,0.f,0.f};

    for (int k0 = 0; k0 < HSZ; k0 += 4) {
        v2f a = *(const v2f*)(arow + k0);
        #pragma unroll
        for (int w = 0; w < 4; ++w) {
            v2f b = *(const v2f*)(brow[w] + k0);
            acc[w] = __builtin_amdgcn_wmma_f32_16x16x4_f32(
                         false, a, false, b, (short)0, acc[w], false, false);
        }
    }

    const int rb = (lane >> 4) << 3;   // 0 or 8
    #pragma unroll
    for (int w = 0; w < 4; ++w) {
        const int c = wave * 64 + w * 16 + m;
        const float bias = Vb[c];
        #pragma unroll
        for (int i = 0; i < 8; ++i)
            lg[(rb + i) * (OSZ + 4) + c] = acc[w][i] + bias;
    }
    __syncthreads();

    // log_softmax: wave w handles rows 2w (lanes 0-15) and 2w+1 (lanes 16-31).
    const int row = 2 * wave + (lane >> 4);
    const float* lr = lg + row * (OSZ + 4);

    float mx = -INFINITY;
    for (int i = m; i < OSZ; i += 16) mx = fmaxf(mx, lr[i]);
    #pragma unroll
    for (int off = 8; off; off >>= 1) mx = fmaxf(mx, __shfl_xor(mx, off, 16));

    float se = 0.f;
    for (int i = m; i < OSZ; i += 16) se += expf(lr[i] - mx);
    #pragma unroll
    for (int off = 8; off; off >>= 1) se += __shfl_xor(se, off, 16);

    const float lse = mx + logf(se);
    for (int i = m; i < OSZ; i += 16)
        out[(size_t)(t0 + row) * OSZ + i] = lr[i] - lse;
}

// ---------------------------------------------------------------------------
// Init: h0 <- hidden input, clear double buffer + barrier counter/flag.
// ---------------------------------------------------------------------------
__global__ void rnn_init(const float* __restrict__ hidden,
                         float* __restrict__ hb,
                         unsigned* __restrict__ counter,
                         unsigned* __restrict__ flag)
{
    const int i = blockIdx.x * blockDim.x + threadIdx.x;
    if (i < HSZ) { hb[i] = hidden[i]; hb[HSZ + i] = 0.f; }
    if (i == 0) { *counter = 0u; *flag = 0u; }
}

// h_final = H[SEQ-1] appended after the [SEQ,OSZ] log-softmax output.
__global__ void copy_hfinal(const float* __restrict__ xuH, float* __restrict__ out)
{
    const int i = blockIdx.x * blockDim.x + threadIdx.x;
    if (i < HSZ)
        out[(size_t)SEQ * OSZ + i] = xuH[(size_t)(SEQ - 1) * HSZ + i];
}

// ---------------------------------------------------------------------------
extern "C" void kernel_launch(void* const* d_in, const int* in_sizes, int n_in,
                              void* d_out, int out_size, void* d_ws, size_t ws_size,
                              hipStream_t stream)
{
    const float* x      = (const float*)d_in[0];  // [SEQ, ISZ]
    const float* hidden = (const float*)d_in[1];  // [1, HSZ]
    const float* Uw     = (const float*)d_in[2];  // [HSZ, ISZ]
    const float* Ub     = (const float*)d_in[3];  // [HSZ]
    const float* Ww     = (const float*)d_in[4];  // [HSZ, HSZ]
    const float* Wb     = (const float*)d_in[5];  // [HSZ]
    const float* Vw     = (const float*)d_in[6];  // [OSZ, HSZ]
    const float* Vb     = (const float*)d_in[7];  // [OSZ]
    float* out = (float*)d_out;

    // Workspace layout: Xu/H [SEQ*HSZ] f32, h double buffer [2*HSZ], counter+flag.
    float*    xuH     = (float*)d_ws;
    float*    hbuf    = xuH + (size_t)SEQ * HSZ;
    unsigned* counter = (unsigned*)(hbuf + 2 * HSZ);
    unsigned* flag    = counter + 1;

    rnn_init<<<(HSZ + 255) / 256, 256, 0, stream>>>(hidden, hbuf, counter, flag);

    dim3 gA(SEQ / 16, HSZ / 128);
    xu_gemm<<<gA, 256, 0, stream>>>(x, Uw, Ub, xuH);

    rnn_recur<<<NWG_B, 256, 0, stream>>>(Ww, Wb, xuH, hbuf, counter, flag);

    out_gemm_lsm<<<SEQ / 16, 256, 0, stream>>>(xuH, Vw, Vb, out);

    copy_hfinal<<<(HSZ + 255) / 256, 256, 0, stream>>>(xuH, out);
}